// PNA_1262720385733
// MI455X (gfx1250) — compile-verified
//
#include <hip/hip_runtime.h>
#include <math.h>

#define Nn 65536
#define Ee 1048576
#define Cc 128
#define Tt 4
#define FIN 32
#define AVG_LOG 1.1211249f

typedef __attribute__((ext_vector_type(2))) float v2f;
typedef __attribute__((ext_vector_type(8))) float v8f;

__device__ __forceinline__ v8f wmma4(v2f a, v2f b, v8f c) {
  // V_WMMA_F32_16X16X4_F32 : D = A(16x4) * B(4x16) + C(16x16)
  return __builtin_amdgcn_wmma_f32_16x16x4_f32(false, a, false, b, (short)0, c,
                                               false, false);
}
__device__ __forceinline__ v2f ld2(const float* p) { return *(const v2f*)p; }

// ---------------- CSR build ----------------
__global__ void k_zero(int* cnt) {
  int i = blockIdx.x * blockDim.x + threadIdx.x;
  if (i < Nn) cnt[i] = 0;
}

__global__ void k_hist(const int* __restrict__ dst, int* __restrict__ cnt) {
  int e = blockIdx.x * blockDim.x + threadIdx.x;
  if (e < Ee) atomicAdd(&cnt[dst[e]], 1);
}

__global__ void k_scan(const int* __restrict__ cnt, int* __restrict__ rowptr,
                       int* __restrict__ cursor) {
  __shared__ int sh[1024];
  __shared__ int carry_s;
  const int tid = threadIdx.x;
  if (tid == 0) carry_s = 0;
  __syncthreads();
  for (int b = 0; b < Nn / 1024; b++) {
    const int i = b * 1024 + tid;
    const int v = cnt[i];
    sh[tid] = v;
    __syncthreads();
    for (int s = 1; s < 1024; s <<= 1) {
      int add = (tid >= s) ? sh[tid - s] : 0;
      __syncthreads();
      sh[tid] += add;
      __syncthreads();
    }
    const int incl = sh[tid];
    const int c = carry_s;
    __syncthreads();
    const int excl = c + incl - v;
    rowptr[i] = excl;
    cursor[i] = excl;
    if (tid == 1023) carry_s = c + incl;
    __syncthreads();
  }
  if (tid == 0) rowptr[Nn] = carry_s;  // == E
}

__global__ void k_scatter(const int* __restrict__ dst, int* __restrict__ cursor,
                          int* __restrict__ csr) {
  int e = blockIdx.x * blockDim.x + threadIdx.x;
  if (e < Ee) {
    int p = atomicAdd(&cursor[dst[e]], 1);
    csr[p] = e;
  }
}

// transpose W_pre (T,96,32)->(T,32,96) and W_post (T,416,32)->(T,32,416)
__global__ void k_transpose(const float* __restrict__ W_pre,
                            const float* __restrict__ W_post,
                            float* __restrict__ W_preT,
                            float* __restrict__ W_postT) {
  int i = blockIdx.x * blockDim.x + threadIdx.x;
  if (i < Tt * FIN * 96) {
    int t = i / (FIN * 96), r = i % (FIN * 96);
    int g = r / 96, k = r % 96;
    W_preT[i] = W_pre[t * 96 * FIN + k * FIN + g];
  } else {
    int j = i - Tt * FIN * 96;
    if (j < Tt * FIN * 416) {
      int t = j / (FIN * 416), r = j % (FIN * 416);
      int g = r / 416, k = r % 416;
      W_postT[j] = W_post[t * 416 * FIN + k * FIN + g];
    }
  }
}

// ---------------- Edge aggregation: 1 node per wave ----------------
__global__ __launch_bounds__(128) void k_edge(
    const float* __restrict__ x, const float* __restrict__ eattr,
    const int* __restrict__ srcI, const float* __restrict__ W_edge,
    const float* __restrict__ b_edge, const float* __restrict__ W_preT,
    const float* __restrict__ b_pre, const int* __restrict__ rowptr,
    const int* __restrict__ csr, float* __restrict__ agg,
    float* __restrict__ cntf) {
  const int wave = threadIdx.x >> 5;
  const int lane = threadIdx.x & 31;
  const int half = lane >> 4;   // lane half selects K pair (A) / K rows (B)
  const int lr = lane & 15;     // A-row (edge in tile) / B-col (feature)
  const int n = blockIdx.x * 4 + wave;

  __shared__ float eproj_s[4][16 * 36];  // 16 rows, stride 36 (pad -> 4r%64)
  float* eproj = eproj_s[wave];

  const int r0 = rowptr[n];
  const int deg = rowptr[n + 1] - r0;

  float s_sum[8], s_sq[8], s_mx[8], s_mn[8];
#pragma unroll
  for (int i = 0; i < 8; i++) {
    s_sum[i] = 0.f; s_sq[i] = 0.f; s_mx[i] = -3.4e38f; s_mn[i] = 3.4e38f;
  }

  const float be0 = b_edge[lr];
  const float be1 = b_edge[16 + lr];
  const float* xn = x + (long)n * Cc;

  for (int base = 0; base < deg; base += 16) {
    int ei = base + lr;
    if (ei > deg - 1) ei = deg - 1;  // clamp padded rows; masked at reduction
    const int e = csr[r0 + ei];
    const long eoff = (long)e * Cc;
    const int srow = srcI[e];
    const float* xs = x + (long)srow * Cc;

    // ---- e_proj = eattr[e] @ W_edge.T + b_edge   (16 edges x 32 out)
    v8f d0 = {}; v8f d1 = {};
    {
      const float* ar = eattr + eoff + 2 * half;
      const float* wb0 = W_edge + (long)lr * Cc + 2 * half;
      const float* wb1 = W_edge + (long)(16 + lr) * Cc + 2 * half;
      for (int k = 0; k < Cc; k += 4) {
        v2f a = ld2(ar + k);
        d0 = wmma4(a, ld2(wb0 + k), d0);
        d1 = wmma4(a, ld2(wb1 + k), d1);
      }
    }
#pragma unroll
    for (int c = 0; c < 8; c++) { d0[c] += be0; d1[c] += be1; }
    // D-layout (lane=N, vgpr=M) -> LDS row-major [M][feat] for A-reads
#pragma unroll
    for (int c = 0; c < 8; c++) {
      eproj[(c + 8 * half) * 36 + lr] = d0[c];
      eproj[(c + 8 * half) * 36 + 16 + lr] = d1[c];
    }
    asm volatile("s_wait_dscnt 0" ::: "memory");

    const int vr = deg - base;  // valid rows in this tile
    for (int t = 0; t < Tt; t++) {
      v8f m0 = {}; v8f m1 = {};
      const float* wp0 = W_preT + (long)(t * FIN + lr) * 96 + 2 * half;
      const float* wp1 = W_preT + (long)(t * FIN + 16 + lr) * 96 + 2 * half;
      const float* xd = xn + t * FIN + 2 * half;   // dst block (bcast rows)
      const float* xr = xs + t * FIN + 2 * half;   // src block (gather)
      const float* ep = eproj + lr * 36 + 2 * half;
#pragma unroll
      for (int k = 0; k < 32; k += 4) {            // h[:, 0:32] = xt[dst]
        v2f a = ld2(xd + k);
        m0 = wmma4(a, ld2(wp0 + k), m0);
        m1 = wmma4(a, ld2(wp1 + k), m1);
      }
#pragma unroll
      for (int k = 0; k < 32; k += 4) {            // h[:,32:64] = xt[src]
        v2f a = ld2(xr + k);
        m0 = wmma4(a, ld2(wp0 + 32 + k), m0);
        m1 = wmma4(a, ld2(wp1 + 32 + k), m1);
      }
#pragma unroll
      for (int k = 0; k < 32; k += 4) {            // h[:,64:96] = e_proj
        v2f a = ld2(ep + k);
        m0 = wmma4(a, ld2(wp0 + 64 + k), m0);
        m1 = wmma4(a, ld2(wp1 + 64 + k), m1);
      }
      const float bp0 = b_pre[t * FIN + lr];
      const float bp1 = b_pre[t * FIN + 16 + lr];
      const int i0 = t * 2, i1 = t * 2 + 1;
#pragma unroll
      for (int c = 0; c < 8; c++) {
        const int M = c + 8 * half;
        const bool ok = (M < vr);
        const float v0 = m0[c] + bp0;
        const float v1 = m1[c] + bp1;
        s_sum[i0] += ok ? v0 : 0.f;
        s_sq[i0] += ok ? v0 * v0 : 0.f;
        s_mx[i0] = ok ? fmaxf(s_mx[i0], v0) : s_mx[i0];
        s_mn[i0] = ok ? fminf(s_mn[i0], v0) : s_mn[i0];
        s_sum[i1] += ok ? v1 : 0.f;
        s_sq[i1] += ok ? v1 * v1 : 0.f;
        s_mx[i1] = ok ? fmaxf(s_mx[i1], v1) : s_mx[i1];
        s_mn[i1] = ok ? fminf(s_mn[i1], v1) : s_mn[i1];
      }
    }
  }

  const float cntv = (float)deg;
  const float safe = fmaxf(cntv, 1.f);
#pragma unroll
  for (int i = 0; i < 8; i++) {
    float sum = s_sum[i] + __shfl_xor(s_sum[i], 16, 32);
    float sq = s_sq[i] + __shfl_xor(s_sq[i], 16, 32);
    float mx = fmaxf(s_mx[i], __shfl_xor(s_mx[i], 16, 32));
    float mn = fminf(s_mn[i], __shfl_xor(s_mn[i], 16, 32));
    float mean = sum / safe;
    float var = sq / safe - mean * mean;
    var = fmaxf(var, 0.f);
    float sd = sqrtf(var + 1e-5f);
    if (deg == 0) { mx = 0.f; mn = 0.f; }
    if (half == 0) {
      const int t = i >> 1, nt = i & 1;
      float* a = agg + (long)n * 512 + t * 128;
      const int f = nt * 16 + lr;
      a[f] = mean; a[32 + f] = mx; a[64 + f] = mn; a[96 + f] = sd;
    }
  }
  if (lane == 0) cntf[n] = cntv;
}

// ---------------- Post: 16 nodes per wave ----------------
__global__ __launch_bounds__(128) void k_post(
    const float* __restrict__ x, const float* __restrict__ agg,
    const float* __restrict__ cntf, const float* __restrict__ W_postT,
    const float* __restrict__ b_post, const float* __restrict__ W_lin,
    const float* __restrict__ b_lin, float* __restrict__ out) {
  const int wave = threadIdx.x >> 5;
  const int lane = threadIdx.x & 31;
  const int half = lane >> 4;
  const int lr = lane & 15;
  const int n0 = (blockIdx.x * 4 + wave) * 16;

  __shared__ float hid_s[4][16 * 132];  // 16 rows, stride 132 dwords
  float* hid = hid_s[wave];

  const int myn = n0 + lr;  // A-row lr <-> node myn
  const float dg = cntf[myn];
  const float logd = logf(dg + 1.f);
  const float amp_s = logd * (1.f / AVG_LOG);
  const float att_s = (dg == 0.f) ? 1.f : (AVG_LOG / logd);

  const float* xrow = x + (long)myn * Cc;
  const float* arow = agg + (long)myn * 512;

  for (int t = 0; t < Tt; t++) {
    v8f o0 = {}; v8f o1 = {};
    const float* wb0 = W_postT + (long)(t * FIN + lr) * 416 + 2 * half;
    const float* wb1 = W_postT + (long)(t * FIN + 16 + lr) * 416 + 2 * half;
    const float* xa = xrow + t * FIN + 2 * half;
    const float* ag = arow + t * 128 + 2 * half;
#pragma unroll
    for (int k = 0; k < 32; k += 4) {        // feat[0:32]   = xt
      v2f a = ld2(xa + k);
      o0 = wmma4(a, ld2(wb0 + k), o0);
      o1 = wmma4(a, ld2(wb1 + k), o1);
    }
    for (int k = 0; k < 128; k += 4) {       // feat[32:160] = agg
      v2f a = ld2(ag + k);
      o0 = wmma4(a, ld2(wb0 + 32 + k), o0);
      o1 = wmma4(a, ld2(wb1 + 32 + k), o1);
    }
    for (int k = 0; k < 128; k += 4) {       // feat[160:288] = agg*amp
      v2f a = ld2(ag + k);
      a.x *= amp_s; a.y *= amp_s;
      o0 = wmma4(a, ld2(wb0 + 160 + k), o0);
      o1 = wmma4(a, ld2(wb1 + 160 + k), o1);
    }
    for (int k = 0; k < 128; k += 4) {       // feat[288:416] = agg*att
      v2f a = ld2(ag + k);
      a.x *= att_s; a.y *= att_s;
      o0 = wmma4(a, ld2(wb0 + 288 + k), o0);
      o1 = wmma4(a, ld2(wb1 + 288 + k), o1);
    }
    const float bp0 = b_post[t * FIN + lr];
    const float bp1 = b_post[t * FIN + 16 + lr];
#pragma unroll
    for (int c = 0; c < 8; c++) {
      hid[(c + 8 * half) * 132 + t * FIN + lr] = o0[c] + bp0;
      hid[(c + 8 * half) * 132 + t * FIN + 16 + lr] = o1[c] + bp1;
    }
  }
  asm volatile("s_wait_dscnt 0" ::: "memory");

  // out = hid(16x128) @ W_lin.T + b_lin + x
  const float* hrow = hid + lr * 132 + 2 * half;
  for (int nt = 0; nt < 8; nt++) {
    v8f acc = {};
    const float* wl = W_lin + (long)(nt * 16 + lr) * Cc + 2 * half;
    for (int k = 0; k < Cc; k += 4) {
      acc = wmma4(ld2(hrow + k), ld2(wl + k), acc);
    }
    const float bl = b_lin[nt * 16 + lr];
#pragma unroll
    for (int c = 0; c < 8; c++) {
      const int row = c + 8 * half;
      const long o = (long)(n0 + row) * Cc + nt * 16 + lr;
      out[o] = acc[c] + bl + x[o];
    }
  }
}

extern "C" void kernel_launch(void* const* d_in, const int* in_sizes, int n_in,
                              void* d_out, int out_size, void* d_ws,
                              size_t ws_size, hipStream_t stream) {
  (void)in_sizes; (void)n_in; (void)out_size; (void)ws_size;
  const float* x      = (const float*)d_in[0];
  const float* eattr  = (const float*)d_in[1];
  const int*   eidx   = (const int*)d_in[2];
  const float* W_edge = (const float*)d_in[3];
  const float* b_edge = (const float*)d_in[4];
  const float* W_pre  = (const float*)d_in[5];
  const float* b_pre  = (const float*)d_in[6];
  const float* W_post = (const float*)d_in[7];
  const float* b_post = (const float*)d_in[8];
  const float* W_lin  = (const float*)d_in[9];
  const float* b_lin  = (const float*)d_in[10];
  float* out = (float*)d_out;
  const int* src = eidx;
  const int* dst = eidx + Ee;

  size_t off = 0;
  char* base = (char*)d_ws;
  auto alloc = [&](size_t bytes) -> void* {
    void* p = base + off;
    off += (bytes + 511) & ~(size_t)511;
    return p;
  };
  int*   cnt_i   = (int*)alloc((size_t)Nn * 4);
  int*   rowptr  = (int*)alloc((size_t)(Nn + 1) * 4);
  int*   cursor  = (int*)alloc((size_t)Nn * 4);
  int*   csr     = (int*)alloc((size_t)Ee * 4);
  float* cntf    = (float*)alloc((size_t)Nn * 4);
  float* W_preT  = (float*)alloc((size_t)Tt * FIN * 96 * 4);
  float* W_postT = (float*)alloc((size_t)Tt * FIN * 416 * 4);
  float* agg     = (float*)alloc((size_t)Nn * 512 * 4);   // (N,T,128)

  k_zero<<<Nn / 256, 256, 0, stream>>>(cnt_i);
  k_hist<<<Ee / 256, 256, 0, stream>>>(dst, cnt_i);
  k_scan<<<1, 1024, 0, stream>>>(cnt_i, rowptr, cursor);
  k_scatter<<<Ee / 256, 256, 0, stream>>>(dst, cursor, csr);
  k_transpose<<<(Tt * FIN * 96 + Tt * FIN * 416) / 256, 256, 0, stream>>>(
      W_pre, W_post, W_preT, W_postT);
  k_edge<<<Nn / 4, 128, 0, stream>>>(x, eattr, src, W_edge, b_edge, W_preT,
                                     b_pre, rowptr, csr, agg, cntf);
  k_post<<<Nn / 64, 128, 0, stream>>>(x, agg, cntf, W_postT, b_post, W_lin,
                                      b_lin, out);
}